// DdlgAutoencoder_90443421319690
// MI455X (gfx1250) — compile-verified
//
#include <hip/hip_runtime.h>
#include <math.h>

#define THREADS 256

typedef unsigned int u32;
typedef u32 u32x4 __attribute__((ext_vector_type(4)));
typedef int i32x8 __attribute__((ext_vector_type(8)));
typedef int i32x4 __attribute__((ext_vector_type(4)));

// --------------------------------------------------------------------------
// Feature detection
// --------------------------------------------------------------------------
#ifdef __has_builtin
#if __has_builtin(__builtin_amdgcn_tensor_load_to_lds) && \
    __has_builtin(__builtin_amdgcn_s_wait_tensorcnt)
#define HAVE_TDM 1
#endif
#if __has_builtin(__builtin_amdgcn_s_wait_asynccnt)
#define WAIT_ASYNC0() __builtin_amdgcn_s_wait_asynccnt(0)
#endif
#endif
#ifndef WAIT_ASYNC0
#define WAIT_ASYNC0() asm volatile("s_wait_asynccnt 0" ::: "memory")
#endif

// Low 32 bits of a generic pointer to a __shared__ object == LDS byte address.
__device__ __forceinline__ unsigned lds_addr32(const void* p) {
    return (unsigned)(unsigned long long)p;
}

#ifdef HAVE_TDM
// One-shot TDM DMA: copy nelem f32 (contiguous) from global to LDS.
// D# per CDNA5 ISA ch.8: group0 = {flags, lds_addr, global_addr, type},
// group1 = dims/strides (1-D tile: tensor_dim0 = tile_dim0 = nelem).
// This toolchain exposes the 6-arg builtin:
//   (uint32x4 g0, int32x8 g1, int32x4 g2, int32x4 g3, int32x8 gx, i32 cpol)
__device__ __forceinline__ void tdm_load_1d(const void* gsrc,
                                            unsigned lds_byte_addr,
                                            unsigned nelem) {
    const unsigned long long ga = (unsigned long long)gsrc;
    u32x4 g0;
    g0.x = 0x1u;                                          // count=1 (valid user D#)
    g0.y = lds_byte_addr;                                 // lds_addr (bytes)
    g0.z = (u32)(ga & 0xFFFFFFFFu);                       // global_addr[31:0]
    g0.w = (u32)((ga >> 32) & 0x01FFFFFFu) | (2u << 30);  // addr[56:32] | type=2
    i32x8 g1;
    g1[0] = (int)(2u << 16);                              // data_size=4B; mask=0
    g1[1] = (int)((nelem & 0xFFFFu) << 16);               // tensor_dim0[15:0] @48
    g1[2] = (int)((nelem >> 16) | (1u << 16));            // dim0[31:16] | tensor_dim1=1
    g1[3] = (int)((nelem & 0xFFFFu) << 16);               // tile_dim0 @112
    g1[4] = 0;                                            // tile_dim1=0, tile_dim2=0
    g1[5] = (int)nelem;                                   // tensor_dim0_stride lo
    g1[6] = 0;                                            // stride0 hi | stride1 lo
    g1[7] = 0;                                            // stride1 hi
    const i32x4 z4 = {0, 0, 0, 0};                        // groups 2/3 unused (<=2D)
    const i32x8 z8 = {0, 0, 0, 0, 0, 0, 0, 0};
    __builtin_amdgcn_tensor_load_to_lds(g0, g1, z4, z4, z8, 0);
}
#endif

// --------------------------------------------------------------------------
// One DDLG layer.  Block = RB batch rows x OT output neurons.  The RB rows
// (contiguous RB*IN floats == 128 KB) are DMA'd into LDS (TDM, or per-lane
// async-to-LDS fallback); gathers then hit LDS.  Each thread owns
// NPT = OT/THREADS adjacent neurons -> coalesced float2 stores.
// --------------------------------------------------------------------------
template <int IN, int RB, int OT>
__global__ __launch_bounds__(THREADS)
void ddlg_layer(const float* __restrict__ x,
                const float* __restrict__ w,
                const int* __restrict__ idx,
                float* __restrict__ out,
                const int OUT,
                const int* __restrict__ is_train_p)
{
    extern __shared__ float tile[];                 // RB * IN floats (128 KB)
    constexpr int NPT = OT / THREADS;               // neurons per thread
    static_assert(NPT == 2, "float2 epilogue assumes 2 neurons/thread");
    constexpr unsigned NELEM = (unsigned)(RB * IN); // 32768 for every layer

    const int b0    = blockIdx.x * RB;
    const int o0    = blockIdx.y * OT;
    const int obase = o0 + (int)threadIdx.x * NPT;

    // ---- kick off the tile DMA ------------------------------------------
    const float* src = x + (size_t)b0 * IN;
#ifdef HAVE_TDM
    if (threadIdx.x < 32) {                         // wave 0 issues one TDM op
        tdm_load_1d(src, lds_addr32(tile), NELEM);
    }
#else
    {
        const unsigned long long ga = (unsigned long long)src;
        const unsigned ldsbase = lds_addr32(tile);
        constexpr int CHUNKS = (int)(NELEM * 4u) / 16;
        for (int c = threadIdx.x; c < CHUNKS; c += THREADS) {
            const unsigned loff = ldsbase + (unsigned)(c * 16);
            const unsigned long long g = ga + (unsigned long long)(c * 16);
            asm volatile("global_load_async_to_lds_b128 %0, %1, off"
                         :: "v"(loff), "v"(g) : "memory");
        }
    }
#endif

    // ---- overlap: per-neuron indices + op-mix probabilities -------------
    const int train = is_train_p[0];
    int4  jA[NPT], jB[NPT];
    float P0[NPT], P1[NPT], P2[NPT], P3[NPT];
#pragma unroll
    for (int n = 0; n < NPT; ++n) {
        const int o = obase + n;
        const int4* jp = (const int4*)(idx + (size_t)o * 8);
        jA[n] = jp[0];
        jB[n] = jp[1];
        const float4 wv = ((const float4*)w)[o];
        if (train) {
            const float m  = fmaxf(fmaxf(wv.x, wv.y), fmaxf(wv.z, wv.w));
            const float e0 = __expf(wv.x - m);
            const float e1 = __expf(wv.y - m);
            const float e2 = __expf(wv.z - m);
            const float e3 = __expf(wv.w - m);
            const float inv = __fdividef(1.0f, e0 + e1 + e2 + e3);
            P0[n] = e0 * inv; P1[n] = e1 * inv;
            P2[n] = e2 * inv; P3[n] = e3 * inv;
        } else {
            int am = 0; float bm = wv.x;
            if (wv.y > bm) { bm = wv.y; am = 1; }
            if (wv.z > bm) { bm = wv.z; am = 2; }
            if (wv.w > bm) { bm = wv.w; am = 3; }
            P0[n] = (am == 0) ? 1.0f : 0.0f;
            P1[n] = (am == 1) ? 1.0f : 0.0f;
            P2[n] = (am == 2) ? 1.0f : 0.0f;
            P3[n] = (am == 3) ? 1.0f : 0.0f;
        }
    }

    // ---- wait for the DMA, then barrier ---------------------------------
#ifdef HAVE_TDM
    if (threadIdx.x < 32) {
        __builtin_amdgcn_s_wait_tensorcnt(0);
    }
#else
    WAIT_ASYNC0();
#endif
    __syncthreads();

    // ---- compute --------------------------------------------------------
    for (int r = 0; r < RB; ++r) {
        const float* row = tile + r * IN;
        float res[NPT];
#pragma unroll
        for (int n = 0; n < NPT; ++n) {
            const int4 ja = jA[n], jb = jB[n];
            const float f0 = row[ja.x], f1 = row[ja.y];
            const float f2 = row[ja.z], f3 = row[ja.w];
            const float f4 = row[jb.x], f5 = row[jb.y];
            const float f6 = row[jb.z], f7 = row[jb.w];

            const float mn = fminf(fminf(fminf(f0, f1), fminf(f2, f3)),
                                   fminf(fminf(f4, f5), fminf(f6, f7)));
            const float mx = fmaxf(fmaxf(fmaxf(f0, f1), fmaxf(f2, f3)),
                                   fmaxf(fmaxf(f4, f5), fmaxf(f6, f7)));

            float a = f0;
#define EIN_STEP(b_) { const float ab_ = a * (b_); \
                       a = __fdividef(ab_, 2.0f - (a + (b_) - ab_)); }
            EIN_STEP(f1) EIN_STEP(f2) EIN_STEP(f3)
            EIN_STEP(f4) EIN_STEP(f5) EIN_STEP(f6) EIN_STEP(f7)
#undef EIN_STEP
            float c = f0;
#define CO_STEP(b_) { c = __fdividef(c + (b_), 1.0f + c * (b_)); }
            CO_STEP(f1) CO_STEP(f2) CO_STEP(f3)
            CO_STEP(f4) CO_STEP(f5) CO_STEP(f6) CO_STEP(f7)
#undef CO_STEP
            res[n] = P0[n] * mn + P1[n] * mx + P2[n] * a + P3[n] * c;
        }
        float2 st; st.x = res[0]; st.y = res[1];
        *(float2*)(out + (size_t)(b0 + r) * OUT + obase) = st;
    }
}

// --------------------------------------------------------------------------
// Launch: 4 chained layers.  4096 -> 2048 -> 1024 -> 2048 -> 4096, B = 4096.
// Workspace: h1 (32MB) @0, h2 (16MB) @32MB, h3 reuses @0 (h1 dead after L2).
// --------------------------------------------------------------------------
extern "C" void kernel_launch(void* const* d_in, const int* in_sizes, int n_in,
                              void* d_out, int out_size, void* d_ws, size_t ws_size,
                              hipStream_t stream)
{
    (void)in_sizes; (void)n_in; (void)out_size; (void)ws_size;

    const float* x  = (const float*)d_in[0];
    const float* w0 = (const float*)d_in[1];
    const float* w1 = (const float*)d_in[2];
    const float* w2 = (const float*)d_in[3];
    const float* w3 = (const float*)d_in[4];
    const int*   i0 = (const int*)d_in[5];
    const int*   i1 = (const int*)d_in[6];
    const int*   i2 = (const int*)d_in[7];
    const int*   i3 = (const int*)d_in[8];
    const int*   tr = (const int*)d_in[9];

    float* h1 = (float*)d_ws;                  // 4096*2048
    float* h2 = h1 + (size_t)4096 * 2048;      // 4096*1024
    float* h3 = (float*)d_ws;                  // reuse (h1 dead after L2)
    float* y  = (float*)d_out;                 // 4096*4096

    const dim3   blk(THREADS);
    const size_t lds = 128u * 1024u;           // RB*IN*4 == 128 KB always

    ddlg_layer<4096,  8, 512><<<dim3(512, 4), blk, lds, stream>>>(x,  w0, i0, h1, 2048, tr);
    ddlg_layer<2048, 16, 512><<<dim3(256, 2), blk, lds, stream>>>(h1, w1, i1, h2, 1024, tr);
    ddlg_layer<1024, 32, 512><<<dim3(128, 4), blk, lds, stream>>>(h2, w2, i2, h3, 2048, tr);
    ddlg_layer<2048, 16, 512><<<dim3(256, 8), blk, lds, stream>>>(h3, w3, i3, y,  4096, tr);
}